// GATNet_40956808135253
// MI455X (gfx1250) — compile-verified
//
#include <hip/hip_runtime.h>
#include <hip/hip_bf16.h>
#include <math.h>

typedef float v2f __attribute__((ext_vector_type(2)));
typedef float v8f __attribute__((ext_vector_type(8)));

#define NEG_SLOPE 0.2f

// ---------------------------------------------------------------------------
// float atomic max via integer compare trick (init value must be -inf)
// ---------------------------------------------------------------------------
__device__ __forceinline__ void atomicMaxFloat(float* addr, float value) {
    if (value >= 0.0f) {
        atomicMax((int*)addr, __float_as_int(value));
    } else {
        atomicMin((unsigned int*)addr, __float_as_uint(value));
    }
}

// ---------------------------------------------------------------------------
// Operand prep (run once per launch; tiny vs. main traffic)
// ---------------------------------------------------------------------------
// Ap[M, Kpad] = A[M, K] zero-padded in K
__global__ void pad_A(const float* __restrict__ A, float* __restrict__ Ap,
                      int M, int K, int Kpad) {
    const int idx = blockIdx.x * blockDim.x + threadIdx.x;
    if (idx >= M * Kpad) return;
    const int r = idx / Kpad, k = idx - r * Kpad;
    Ap[idx] = (k < K) ? A[(size_t)r * K + k] : 0.0f;
}

// Pair-interleaved, padded weight pack:
//   Bp[((k>>1)*Npad + n)*2 + (k&1)] = B[k, n]  (zero outside K x N)
// so one lane's B fragment {B[k][n], B[k+1][n]} is one contiguous b64 load.
__global__ void pack_B(const float* __restrict__ B, float* __restrict__ Bp,
                       int K, int N, int Kpad, int Npad) {
    const int idx = blockIdx.x * blockDim.x + threadIdx.x;
    if (idx >= Kpad * Npad) return;
    const int k = idx / Npad, n = idx - k * Npad;
    const float v = (k < K && n < N) ? B[(size_t)k * N + n] : 0.0f;
    Bp[((size_t)(k >> 1) * Npad + n) * 2 + (k & 1)] = v;
}

// ---------------------------------------------------------------------------
// GEMM core: C[M, Npad] = A[M, Kpad] * B[Kpad, Npad], V_WMMA_F32_16X16X4_F32.
// Preconditions: M%16==0, Kpad%4==0, Npad%16==0, B pre-packed (pack_B).
// One wave computes one 16x16 C tile; branch-free inner loop:
//   global_load_b64 (A pair) + global_load_b64 (B pair) + v_wmma.
// ISA operand layouts (CDNA5 ISA 7.12.2):
//   A 16x4 : lanes 0-15 -> row=lane, {K0,K1}; lanes 16-31 -> {K2,K3}
//   B 4x16 : lanes 0-15 -> col=lane, {K0,K1}; lanes 16-31 -> {K2,K3}
//   C/D    : vgpr i -> row i (lanes 0-15) / row 8+i (lanes 16-31), col=lane&15
// ---------------------------------------------------------------------------
template <int KPAD, int NPAD>
__device__ __forceinline__ void gemm_body(const float* __restrict__ A,
                                          const float* __restrict__ Bp,
                                          float* __restrict__ C,
                                          int M, int kpadRT, int npadRT) {
    const int kpad = (KPAD > 0) ? KPAD : kpadRT;
    const int npad = (NPAD > 0) ? NPAD : npadRT;

    const int wave  = threadIdx.x >> 5;
    const int lane  = threadIdx.x & 31;
    const int mtile = blockIdx.y * 4 + wave;   // wave-uniform
    if (mtile * 16 >= M) return;               // uniform exit: EXEC stays all-ones
    const int row0   = mtile * 16;
    const int col0   = blockIdx.x * 16;
    const int lane15 = lane & 15;
    const int khalf  = (lane >> 4) << 1;       // 0 or 2

    const float* ap = A + (size_t)(row0 + lane15) * kpad + khalf;
    const float* bp = Bp + ((size_t)(khalf >> 1) * npad + (col0 + lane15)) * 2;
    const size_t bstep = (size_t)npad * 4;     // floats: 2 pair-rows per K-step

    v8f acc = {0.f, 0.f, 0.f, 0.f, 0.f, 0.f, 0.f, 0.f};

    const int steps = kpad >> 2;
    for (int s = 0; s < steps; ++s) {
        const v2f av = *(const v2f*)(ap + (size_t)s * 4);
        const v2f bv = *(const v2f*)(bp + (size_t)s * bstep);
        acc = __builtin_amdgcn_wmma_f32_16x16x4_f32(
                  false, av, false, bv, (short)0, acc, false, false);
    }

    const int col   = col0 + lane15;
    const int rbase = row0 + ((lane >> 4) << 3);
#pragma unroll
    for (int i = 0; i < 8; ++i)
        C[(size_t)(rbase + i) * npad + col] = acc[i];
}

template <int KPAD, int NPAD>
__global__ void gemm_wmma_fixed(const float* __restrict__ A,
                                const float* __restrict__ Bp,
                                float* __restrict__ C, int M) {
    gemm_body<KPAD, NPAD>(A, Bp, C, M, KPAD, NPAD);
}

__global__ void gemm_wmma_generic(const float* __restrict__ A,
                                  const float* __restrict__ Bp,
                                  float* __restrict__ C,
                                  int M, int Kpad, int Npad) {
    gemm_body<0, 0>(A, Bp, C, M, Kpad, Npad);
}

// ---------------------------------------------------------------------------
// Per-node attention score projections: ss[n] = h[n,:].a_src, sd[n] = h[n,:].a_dst
// One wave per node (wave32 shuffle reduction). h has leading dimension ld.
// ---------------------------------------------------------------------------
__global__ void node_scores(const float* __restrict__ h, int ld,
                            const float* __restrict__ a_s,
                            const float* __restrict__ a_d,
                            float* __restrict__ ss, float* __restrict__ sd,
                            int N, int F) {
    const int wid  = (int)((blockIdx.x * blockDim.x + threadIdx.x) >> 5);
    const int lane = threadIdx.x & 31;
    if (wid >= N) return;
    float accs = 0.f, accd = 0.f;
    for (int f = lane; f < F; f += 32) {
        const float hv = h[(size_t)wid * ld + f];
        accs += hv * a_s[f];
        accd += hv * a_d[f];
    }
    for (int off = 16; off > 0; off >>= 1) {
        accs += __shfl_down(accs, off, 32);
        accd += __shfl_down(accd, off, 32);
    }
    if (lane == 0) { ss[wid] = accs; sd[wid] = accd; }
}

// ---------------------------------------------------------------------------
// Layer-3 matvec (Fout = 1) fused with score projection. X: compact [N, K].
// ---------------------------------------------------------------------------
__global__ void matvec_scores(const float* __restrict__ X,
                              const float* __restrict__ w,
                              const float* __restrict__ a_s,
                              const float* __restrict__ a_d,
                              float* __restrict__ h3,
                              float* __restrict__ ss, float* __restrict__ sd,
                              int N, int K) {
    const int wid  = (int)((blockIdx.x * blockDim.x + threadIdx.x) >> 5);
    const int lane = threadIdx.x & 31;
    if (wid >= N) return;
    float acc = 0.f;
    for (int f = lane; f < K; f += 32) acc += X[(size_t)wid * K + f] * w[f];
    for (int off = 16; off > 0; off >>= 1) acc += __shfl_down(acc, off, 32);
    if (lane == 0) {
        h3[wid] = acc;
        ss[wid] = acc * a_s[0];
        sd[wid] = acc * a_d[0];
    }
}

// ---------------------------------------------------------------------------
// Edge list prep: concat(edge_index, self-loops) as int32.
// ---------------------------------------------------------------------------
__global__ void prep_edges(const int* __restrict__ ei,
                           int* __restrict__ src, int* __restrict__ dst,
                           int E, int N) {
    const int e = blockIdx.x * blockDim.x + threadIdx.x;
    if (e >= E + N) return;
    if (e < E) { src[e] = ei[e]; dst[e] = ei[E + e]; }
    else       { const int n = e - E; src[e] = n; dst[e] = n; }
}

__global__ void fill_f32(float* __restrict__ p, float v, int n) {
    const int i = blockIdx.x * blockDim.x + threadIdx.x;
    if (i < n) p[i] = v;
}

// Pass A: leaky-relu logits + segment max (float atomic max)
__global__ void edge_logits(const int* __restrict__ src, const int* __restrict__ dst,
                            const float* __restrict__ ss, const float* __restrict__ sd,
                            float* __restrict__ ebuf, float* __restrict__ m, int Etot) {
    const int e = blockIdx.x * blockDim.x + threadIdx.x;
    if (e >= Etot) return;
    float v = ss[src[e]] + sd[dst[e]];
    v = (v >= 0.f) ? v : NEG_SLOPE * v;
    ebuf[e] = v;
    atomicMaxFloat(&m[dst[e]], v);
}

// Pass B: exp(e - m[dst]) + segment sum
__global__ void edge_exp(const int* __restrict__ dst,
                         float* __restrict__ ebuf,
                         const float* __restrict__ m,
                         float* __restrict__ ssum, int Etot) {
    const int e = blockIdx.x * blockDim.x + threadIdx.x;
    if (e >= Etot) return;
    const int d = dst[e];
    const float w = expf(ebuf[e] - m[d]);
    ebuf[e] = w;
    atomicAdd(&ssum[d], w);
}

// Pass C: agg[dst,:] += alpha * h[src,:]  (wave per edge; h stride ld, agg compact F)
__global__ void edge_scatter(const int* __restrict__ src, const int* __restrict__ dst,
                             const float* __restrict__ w, const float* __restrict__ ssum,
                             const float* __restrict__ h, int ld, float* __restrict__ agg,
                             int Etot, int F) {
    const int wid  = (int)((blockIdx.x * blockDim.x + threadIdx.x) >> 5);
    const int lane = threadIdx.x & 31;
    if (wid >= Etot) return;
    const int s = src[wid], d = dst[wid];
    const float alpha = w[wid] / ssum[d];
    for (int f = lane; f < F; f += 32) {
        atomicAdd(&agg[(size_t)d * F + f], h[(size_t)s * ld + f] * alpha);
    }
}

// Pass C specialised for F == 1 (thread per edge)
__global__ void edge_scatter_f1(const int* __restrict__ src, const int* __restrict__ dst,
                                const float* __restrict__ w, const float* __restrict__ ssum,
                                const float* __restrict__ h, float* __restrict__ agg,
                                int Etot) {
    const int e = blockIdx.x * blockDim.x + threadIdx.x;
    if (e >= Etot) return;
    const int d = dst[e];
    atomicAdd(&agg[d], h[src[e]] * (w[e] / ssum[d]));
}

// out = (relu?) (agg + b)   (out may alias agg; compact [N, F])
__global__ void finalize_bias(const float* __restrict__ agg, const float* __restrict__ b,
                              float* __restrict__ out, int N, int F, int do_relu) {
    const int i = blockIdx.x * blockDim.x + threadIdx.x;
    if (i >= N * F) return;
    float v = agg[i] + b[i % F];
    if (do_relu) v = fmaxf(v, 0.f);
    out[i] = v;
}

// ---------------------------------------------------------------------------
// Host side
// ---------------------------------------------------------------------------
extern "C" void kernel_launch(void* const* d_in, const int* in_sizes, int n_in,
                              void* d_out, int out_size, void* d_ws, size_t ws_size,
                              hipStream_t stream) {
    const float* x   = (const float*)d_in[0];
    const int*   ei  = (const int*)  d_in[1];
    const float* W1  = (const float*)d_in[2];
    const float* a1s = (const float*)d_in[3];
    const float* a1d = (const float*)d_in[4];
    const float* b1  = (const float*)d_in[5];
    const float* W2  = (const float*)d_in[6];
    const float* a2s = (const float*)d_in[7];
    const float* a2d = (const float*)d_in[8];
    const float* b2  = (const float*)d_in[9];
    const float* W3  = (const float*)d_in[10];
    const float* a3s = (const float*)d_in[11];
    const float* a3d = (const float*)d_in[12];
    const float* b3  = (const float*)d_in[13];

    const int F1   = in_sizes[3];            // 300
    const int F2   = in_sizes[7];            // 100
    const int F0   = in_sizes[2] / F1;       // 58
    const int N    = in_sizes[0] / F0;       // 30000
    const int E    = in_sizes[1] / 2;        // 240000
    const int Etot = E + N;                  // 270000
    const int K3   = in_sizes[10];           // 100

    const int K0p = (F0 + 3)  & ~3;          // 60
    const int N1p = (F1 + 15) & ~15;         // 304
    const int K2p = (F1 + 3)  & ~3;          // 300 (already aligned)
    const int N2p = (F2 + 15) & ~15;         // 112

    // ---- workspace carve-out (256B aligned) ----
    char* wp = (char*)d_ws;
    auto carve = [&](size_t bytes) {
        void* r = (void*)wp;
        wp += (bytes + 255) & ~(size_t)255;
        return r;
    };
    int*   srcv   = (int*)  carve((size_t)Etot * 4);
    int*   dstv   = (int*)  carve((size_t)Etot * 4);
    float* xpad   = (float*)carve((size_t)N * K0p * 4);
    float* W1p    = (float*)carve((size_t)K0p * N1p * 4);
    float* W2p    = (float*)carve((size_t)K2p * N2p * 4);
    float* h1     = (float*)carve((size_t)N * N1p * 4);   // ld = N1p
    float* agg1   = (float*)carve((size_t)N * F1 * 4);    // compact
    float* h2     = (float*)carve((size_t)N * N2p * 4);   // ld = N2p
    float* agg2   = (float*)carve((size_t)N * F2 * 4);    // compact
    float* h3     = (float*)carve((size_t)N * 4);
    float* agg3   = (float*)carve((size_t)N * 4);
    float* ss     = (float*)carve((size_t)N * 4);
    float* sd     = (float*)carve((size_t)N * 4);
    float* mbuf   = (float*)carve((size_t)N * 4);
    float* sumbuf = (float*)carve((size_t)N * 4);
    float* ebuf   = (float*)carve((size_t)Etot * 4);
    (void)ws_size; (void)n_in; (void)out_size;

    const float NEG_INF = -__builtin_huge_valf();
    const int BT = 256;
    const int edgeBlocks  = (Etot + BT - 1) / BT;
    const int waveNBlocks = (N * 32 + BT - 1) / BT;
    const int waveEBlocks = (int)(((size_t)Etot * 32 + BT - 1) / BT);

    // ---- prep: edges, padded A, packed weights ----
    prep_edges<<<edgeBlocks, BT, 0, stream>>>(ei, srcv, dstv, E, N);
    pad_A<<<((size_t)N * K0p + BT - 1) / BT, BT, 0, stream>>>(x, xpad, N, F0, K0p);
    pack_B<<<((size_t)K0p * N1p + BT - 1) / BT, BT, 0, stream>>>(W1, W1p, F0, F1, K0p, N1p);
    pack_B<<<((size_t)K2p * N2p + BT - 1) / BT, BT, 0, stream>>>(W2, W2p, F1, F2, K2p, N2p);

    // ---- shared softmax/aggregate pipeline for one GAT layer ----
    auto attention = [&](const float* h, int ldH, float* agg, const float* bias,
                         float* outp, int F, int relu, bool f1) {
        fill_f32<<<(N + BT - 1) / BT, BT, 0, stream>>>(mbuf, NEG_INF, N);
        fill_f32<<<(N + BT - 1) / BT, BT, 0, stream>>>(sumbuf, 0.f, N);
        fill_f32<<<((size_t)N * F + BT - 1) / BT, BT, 0, stream>>>(agg, 0.f, N * F);
        edge_logits<<<edgeBlocks, BT, 0, stream>>>(srcv, dstv, ss, sd, ebuf, mbuf, Etot);
        edge_exp<<<edgeBlocks, BT, 0, stream>>>(dstv, ebuf, mbuf, sumbuf, Etot);
        if (f1)
            edge_scatter_f1<<<edgeBlocks, BT, 0, stream>>>(srcv, dstv, ebuf, sumbuf, h, agg, Etot);
        else
            edge_scatter<<<waveEBlocks, BT, 0, stream>>>(srcv, dstv, ebuf, sumbuf, h, ldH, agg, Etot, F);
        finalize_bias<<<((size_t)N * F + BT - 1) / BT, BT, 0, stream>>>(agg, bias, outp, N, F, relu);
    };

    // ================= Layer 1: [N,60] @ [60,304] =================
    {
        dim3 grid(N1p / 16, (N / 16 + 3) / 4);
        if (K0p == 60 && N1p == 304)
            gemm_wmma_fixed<60, 304><<<grid, 128, 0, stream>>>(xpad, W1p, h1, N);
        else
            gemm_wmma_generic<<<grid, 128, 0, stream>>>(xpad, W1p, h1, N, K0p, N1p);
        node_scores<<<waveNBlocks, BT, 0, stream>>>(h1, N1p, a1s, a1d, ss, sd, N, F1);
        attention(h1, N1p, agg1, b1, agg1, F1, 1, false);
    }
    // ================= Layer 2: [N,300] @ [300,112] =================
    {
        dim3 grid(N2p / 16, (N / 16 + 3) / 4);
        if (K2p == 300 && N2p == 112)
            gemm_wmma_fixed<300, 112><<<grid, 128, 0, stream>>>(agg1, W2p, h2, N);
        else
            gemm_wmma_generic<<<grid, 128, 0, stream>>>(agg1, W2p, h2, N, K2p, N2p);
        node_scores<<<waveNBlocks, BT, 0, stream>>>(h2, N2p, a2s, a2d, ss, sd, N, F2);
        attention(h2, N2p, agg2, b2, agg2, F2, 1, false);
    }
    // ================= Layer 3: [N,100] @ [100,1] =================
    {
        matvec_scores<<<waveNBlocks, BT, 0, stream>>>(agg2, W3, a3s, a3d, h3, ss, sd, N, K3);
        attention(h3, 1, agg3, b3, (float*)d_out, 1, 0, true);
    }
}